// GCN_83897891160159
// MI455X (gfx1250) — compile-verified
//
#include <hip/hip_runtime.h>

typedef float v2f __attribute__((ext_vector_type(2)));
typedef float v8f __attribute__((ext_vector_type(8)));

#define Bb 2
#define Nn 8192
#define Ff 64
#define Oo 64
#define ROWS (Bb * Nn) /* 16384 */

// workspace layout (float offsets)
#define WS_SI    0        /* 16384 : si[b,n]              */
#define WS_SJ    16384    /* 16384 : sj[b,n]              */
#define WS_DIS   32768    /* 16384 : dis[b,n]             */
#define WS_AV    49152    /* 16384 : dis*(si+c)           */
#define WS_PARTS 65536    /*  2048 : per-block sj partials*/
#define WS_S     67584    /*     2 : S[b]                 */
#define WS_U0P   67600    /* 16384 : u0 block partials    */
#define WS_U1P   83984    /* 16384 : u1 block partials    */
#define WS_T     100368   /*   256 : [b*128 + {t0|t1}*64 + o] */

// ---- pass 1: per-row dots si = x.w_i, sj = x.w_j ; partial sum of sj ----
__global__ void k_rows(const float* __restrict__ x, const float* __restrict__ adj_w,
                       float* __restrict__ ws) {
  int lane = threadIdx.x & 31, wave = threadIdx.x >> 5;
  int row = blockIdx.x * 8 + wave;
  float2 xv = ((const float2*)x)[row * 32 + lane];          // x[row][2*lane..2*lane+1]
  float sj = xv.x * adj_w[2 * lane]      + xv.y * adj_w[2 * lane + 1];
  float si = xv.x * adj_w[64 + 2 * lane] + xv.y * adj_w[64 + 2 * lane + 1];
  #pragma unroll
  for (int off = 16; off > 0; off >>= 1) {
    sj += __shfl_xor(sj, off, 32);
    si += __shfl_xor(si, off, 32);
  }
  __shared__ float lsj[8];
  if (lane == 0) {
    ws[WS_SI + row] = si;
    ws[WS_SJ + row] = sj;
    lsj[wave] = sj;
  }
  __syncthreads();
  if (threadIdx.x == 0) {
    float s = 0.f;
    #pragma unroll
    for (int i = 0; i < 8; ++i) s += lsj[i];
    ws[WS_PARTS + blockIdx.x] = s;                          // deterministic partials
  }
}

// ---- pass 2: S[b] = sum of 1024 partials per batch (fixed-order tree) ----
__global__ void k_sumS(float* __restrict__ ws) {
  int b = blockIdx.x, t = threadIdx.x;
  __shared__ float l[256];
  float s = 0.f;
  for (int i = t; i < 1024; i += 256) s += ws[WS_PARTS + b * 1024 + i];
  l[t] = s;
  __syncthreads();
  for (int off = 128; off > 0; off >>= 1) {
    if (t < off) l[t] += l[t + off];
    __syncthreads();
  }
  if (t == 0) ws[WS_S + b] = l[0];
}

// ---- pass 3: dis = rsqrt(max(N*si + S + N*c, 1)); avec = dis*(si+c) ----
__global__ void k_dis(const float* __restrict__ adj_b, float* __restrict__ ws) {
  int idx = blockIdx.x * 256 + threadIdx.x;                 // 0..16383
  int b = idx >> 13;
  float c = adj_b[0];
  float si = ws[WS_SI + idx];
  float deg = (float)Nn * si + ws[WS_S + b] + (float)Nn * c;
  float dis = rsqrtf(fmaxf(deg, 1.0f));
  ws[WS_DIS + idx] = dis;
  ws[WS_AV + idx] = dis * (si + c);
}

// ---- pass 4: u0[k] = sum_r dis[r]*x[r][k] ; u1[k] = sum_r dis[r]*sj[r]*x[r][k] ----
// block = 64 rows, thread (rg,k); coalesced 256B row reads; fixed-order partials.
__global__ void k_upart(const float* __restrict__ x, float* __restrict__ ws) {
  int t = threadIdx.x;
  int k = t & 63, rg = t >> 6;
  int r0 = blockIdx.x * 64;                                 // flat rows, batch at blk 128
  float a0 = 0.f, a1 = 0.f;
  #pragma unroll 4
  for (int i = 0; i < 16; ++i) {
    int r = r0 + i * 4 + rg;
    float d = ws[WS_DIS + r];
    float dsj = d * ws[WS_SJ + r];
    float xv = x[r * 64 + k];
    a0 += d * xv;
    a1 += dsj * xv;
  }
  __shared__ float l0[256], l1[256];
  l0[t] = a0; l1[t] = a1;
  __syncthreads();
  if (t < 64) {
    ws[WS_U0P + blockIdx.x * 64 + t] = l0[t] + l0[64 + t] + l0[128 + t] + l0[192 + t];
    ws[WS_U1P + blockIdx.x * 64 + t] = l1[t] + l1[64 + t] + l1[128 + t] + l1[192 + t];
  }
}

// ---- pass 5: reduce u partials; t{0,1}[b,o] = sum_k u{0,1}[b,k] * W[k,o] ----
__global__ void k_t(const float* __restrict__ weight, float* __restrict__ ws) {
  __shared__ float u0s[128], u1s[128];
  int t = threadIdx.x;
  if (t < 128) {
    int b = t >> 6, k = t & 63;
    float s0 = 0.f, s1 = 0.f;
    for (int i = 0; i < 128; ++i) {
      s0 += ws[WS_U0P + (b * 128 + i) * 64 + k];
      s1 += ws[WS_U1P + (b * 128 + i) * 64 + k];
    }
    u0s[t] = s0; u1s[t] = s1;
  }
  __syncthreads();
  int b = t >> 7, rem = t & 127, which = rem >> 6, o = rem & 63;
  const float* u = which ? u1s : u0s;
  float acc = 0.f;
  for (int k = 0; k < 64; ++k) acc += u[b * 64 + k] * weight[k * 64 + o];
  ws[WS_T + b * 128 + which * 64 + o] = acc;
}

// ---- pass 6: out = relu([avec|dis] @ [t0;t1] + bias) via V_WMMA_F32_16X16X4_F32 ----
// A 16x4 layout: lanes 0-15 carry K=0,1 in the two A VGPRs; lanes 16-31 carry K=2,3 (zero).
// B 4x16 mirrored: lanes 0-15 rows K=0 (t0) / K=1 (t1); lanes 16-31 rows 2,3 (zero).
__global__ void k_final(const float* __restrict__ bias, const float* __restrict__ ws,
                        float* __restrict__ out) {
  int lane = threadIdx.x & 31, wave = threadIdx.x >> 5;
  int rowTile = wave >> 2, colTile = wave & 3;
  int row0 = blockIdx.x * 32 + rowTile * 16;
  int col0 = colTile * 16;
  int b = row0 >> 13;
  int m = lane & 15;
  bool upper = lane >= 16;

  float av = ws[WS_AV + row0 + m];
  float dv = ws[WS_DIS + row0 + m];
  float t0 = ws[WS_T + b * 128 + col0 + m];
  float t1 = ws[WS_T + b * 128 + 64 + col0 + m];

  v2f A, Bv;
  A.x  = upper ? 0.f : av;  A.y  = upper ? 0.f : dv;   // K=2,3 half zeroed
  Bv.x = upper ? 0.f : t0;  Bv.y = upper ? 0.f : t1;
  float cv = bias[col0 + m];
  v8f C = {cv, cv, cv, cv, cv, cv, cv, cv};            // bias depends only on column

  v8f D = __builtin_amdgcn_wmma_f32_16x16x4_f32(
      /*neg_a=*/false, A, /*neg_b=*/false, Bv,
      /*c_mod=*/(short)0, C, /*reuse_a=*/false, /*reuse_b=*/false);

  int colg = col0 + m;
  int rbase = row0 + (upper ? 8 : 0);
  #pragma unroll
  for (int v = 0; v < 8; ++v)
    out[(rbase + v) * 64 + colg] = fmaxf(D[v], 0.0f);
}

extern "C" void kernel_launch(void* const* d_in, const int* in_sizes, int n_in,
                              void* d_out, int out_size, void* d_ws, size_t ws_size,
                              hipStream_t stream) {
  const float* x      = (const float*)d_in[0];
  const float* adj_w  = (const float*)d_in[1];
  const float* adj_b  = (const float*)d_in[2];
  const float* weight = (const float*)d_in[3];
  const float* bias   = (const float*)d_in[4];
  float* ws  = (float*)d_ws;
  float* out = (float*)d_out;

  k_rows <<<ROWS / 8,  256, 0, stream>>>(x, adj_w, ws);
  k_sumS <<<Bb,        256, 0, stream>>>(ws);
  k_dis  <<<ROWS/256,  256, 0, stream>>>(adj_b, ws);
  k_upart<<<ROWS / 64, 256, 0, stream>>>(x, ws);
  k_t    <<<1,         256, 0, stream>>>(weight, ws);
  k_final<<<ROWS / 32, 256, 0, stream>>>(bias, ws, out);
}